// ROIHead_26242250179154
// MI455X (gfx1250) — compile-verified
//
#include <hip/hip_runtime.h>
#include <math.h>

// ---------------- constants (match reference setup) ----------------
#define CFEAT 256
#define HF 50
#define WF 50
#define POOLK 7
#define FDIM (CFEAT * POOLK * POOLK)   // 12544
#define FC 1024
#define NCLS 21
#define SORTN 8192
#define NSMAX 8                        // 8*1024 >= max candidates
#define KC 32                          // GEMM K-chunk
#define LDSW (KC + 4)                  // 36-dword row stride: 16B-aligned rows, conflict-free b64 frags
#define AROWS 32
#define BROWS 64
#define BUFSZ ((AROWS + BROWS) * LDSW) // floats per buffer (A region then B region)
#define BREG (AROWS * LDSW)            // B region offset (floats)

typedef float v2f __attribute__((ext_vector_type(2)));
typedef float v8f __attribute__((ext_vector_type(8)));

// CDNA5 async global->LDS copy, 16B per lane, tracked by ASYNCcnt.
__device__ __forceinline__ void async_copy16(unsigned lds_addr, const float* gptr) {
  asm volatile("global_load_async_to_lds_b128 %0, %1, off"
               :: "v"(lds_addr), "v"(gptr)
               : "memory");
}
__device__ __forceinline__ void wait_async0() {
  asm volatile("s_wait_asynccnt 0" ::: "memory");
}

// ---------------- ROI max pool (torchvision roi_pool semantics) ----------------
__global__ __launch_bounds__(256) void roi_pool_kernel(
    const float* __restrict__ feat, const float* __restrict__ props,
    const int* __restrict__ image_h_ptr, float* __restrict__ X, int Nroi) {
  int roi = blockIdx.x;
  if (roi >= Nroi) {  // zero-pad rows up to Mp
    for (int t = threadIdx.x; t < FDIM; t += 256) X[(size_t)roi * FDIM + t] = 0.0f;
    return;
  }
  float ih = (float)image_h_ptr[0];
  float scale = exp2f(rintf(log2f((float)HF / ih)));  // 2^round(log2(Hf/H))
  float x1 = rintf(props[roi * 4 + 0] * scale);
  float y1 = rintf(props[roi * 4 + 1] * scale);
  float x2 = rintf(props[roi * 4 + 2] * scale);
  float y2 = rintf(props[roi * 4 + 3] * scale);
  float rw = fmaxf(x2 - x1 + 1.0f, 1.0f), rh = fmaxf(y2 - y1 + 1.0f, 1.0f);
  float bw = rw / POOLK, bh = rh / POOLK;
  int wsb[POOLK], web[POOLK], hsb[POOLK], heb[POOLK];
#pragma unroll
  for (int p = 0; p < POOLK; p++) {
    wsb[p] = (int)fminf(fmaxf(floorf(p * bw) + x1, 0.0f), (float)WF);
    web[p] = (int)fminf(fmaxf(ceilf((p + 1) * bw) + x1, 0.0f), (float)WF);
    hsb[p] = (int)fminf(fmaxf(floorf(p * bh) + y1, 0.0f), (float)HF);
    heb[p] = (int)fminf(fmaxf(ceilf((p + 1) * bh) + y1, 0.0f), (float)HF);
  }
  for (int t = threadIdx.x; t < FDIM; t += 256) {
    int c = t / (POOLK * POOLK);
    int b = t % (POOLK * POOLK);
    int ph = b / POOLK, pw = b % POOLK;
    const float* fc = feat + (size_t)c * HF * WF;
    float m = -1e30f;
    bool any = false;
    for (int y = hsb[ph]; y < heb[ph]; y++)
      for (int x = wsb[pw]; x < web[pw]; x++) {
        m = fmaxf(m, fc[y * WF + x]);
        any = true;
      }
    X[(size_t)roi * FDIM + t] = any ? m : 0.0f;  // empty bin -> 0
  }
}

// ---------------- fp32 WMMA GEMM: C[Mp,Nout] = A[Mp,K] @ W[Nout,K]^T + bias ----------------
// grid: (ceil(Nout/64), Mp/32), block 256 (8 waves). Wave -> 16x16 tile via V_WMMA_F32_16X16X4_F32.
// Double-buffered LDS; staging = 3x global_load_async_to_lds_b128 per thread per K-chunk.
__global__ __launch_bounds__(256) void gemm_wmma_kernel(
    const float* __restrict__ A, const float* __restrict__ W,
    const float* __restrict__ bias, float* __restrict__ C,
    int K, int Nout, int relu) {
  __shared__ __align__(16) float smem[2][BUFSZ];
  const int tid = threadIdx.x;
  const int wave = tid >> 5;
  const int lane = tid & 31;
  const int l = lane & 15;
  const int half = lane >> 4;
  const int m_base = blockIdx.y * 32 + (wave >> 2) * 16;
  const int n_blk = blockIdx.x * 64;
  const int n_base = n_blk + (wave & 3) * 16;

  // zero-fill once when the B tile has out-of-range rows (small heads)
  if ((Nout & 63) != 0) {
    for (int t = tid; t < 2 * BUFSZ; t += 256) smem[0][t] = 0.0f;
    __syncthreads();
  }

  // per-thread staging addresses (bump by KC floats per chunk)
  const int srow = tid >> 3;          // 0..31
  const int scol = (tid & 7) * 4;     // 0,4,...,28
  const float* gA  = A + (size_t)(blockIdx.y * 32 + srow) * K + scol;
  const float* gB0 = W + (size_t)(n_blk + srow) * K + scol;
  const float* gB1 = W + (size_t)(n_blk + 32 + srow) * K + scol;
  const int vB0 = (n_blk + srow) < Nout;
  const int vB1 = (n_blk + 32 + srow) < Nout;
  const unsigned base = (unsigned)(uintptr_t)&smem[0][0];
  const unsigned aoff  = base + (unsigned)((srow * LDSW + scol) * 4);
  const unsigned boff0 = base + (unsigned)((BREG + srow * LDSW + scol) * 4);
  const unsigned boff1 = base + (unsigned)((BREG + (32 + srow) * LDSW + scol) * 4);
  const unsigned bufstride = (unsigned)(BUFSZ * 4);

  const int arow = (wave >> 2) * 16 + l;          // A fragment row (M)
  const int brow = (wave & 3) * 16 + l;           // B fragment row (N)

  // prologue: stage chunk 0 into buffer 0
  async_copy16(aoff, gA);
  if (vB0) async_copy16(boff0, gB0);
  if (vB1) async_copy16(boff1, gB1);

  v8f acc = {};
  int buf = 0;
  for (int k0 = 0; k0 < K; k0 += KC) {
    wait_async0();     // my async writes for this buffer are done
    __syncthreads();   // everyone's are visible
    // stage next chunk into the other buffer (its readers finished last iteration)
    if (k0 + KC < K) {
      unsigned nb = (unsigned)(buf ^ 1) * bufstride;
      async_copy16(aoff + nb, gA + k0 + KC);
      if (vB0) async_copy16(boff0 + nb, gB0 + k0 + KC);
      if (vB1) async_copy16(boff1 + nb, gB1 + k0 + KC);
    }
    const float* As_ = &smem[buf][arow * LDSW];
    const float* Bs_ = &smem[buf][BREG + brow * LDSW];
#pragma unroll
    for (int kk = 0; kk < KC; kk += 4) {
      // A 16x4 frag: lane l holds (M=l, K=kk+2*half, kk+2*half+1)
      v2f a = *(const v2f*)&As_[kk + 2 * half];
      // B 4x16 frag: lane l holds (N=l, K=kk+2*half, kk+2*half+1)
      v2f b = *(const v2f*)&Bs_[kk + 2 * half];
      acc = __builtin_amdgcn_wmma_f32_16x16x4_f32(
          false, a, false, b, (short)0, acc, false, false);
    }
    buf ^= 1;
  }
  int n = n_base + l;
  if (n < Nout) {
    float bv = bias[n];
#pragma unroll
    for (int r = 0; r < 8; r++) {  // D VGPR r -> row m_base + r + 8*half
      float v = acc[r] + bv;
      if (relu) v = fmaxf(v, 0.0f);
      C[(size_t)(m_base + r + 8 * half) * Nout + n] = v;
    }
  }
}

// ---------------- softmax + box decode + candidate build ----------------
__global__ __launch_bounds__(256) void cand_kernel(
    const float* __restrict__ cls, const float* __restrict__ reg,
    const float* __restrict__ props, const int* __restrict__ ihp,
    const int* __restrict__ iwp, int Nroi,
    float* __restrict__ nbox, float* __restrict__ obox, float* __restrict__ sc,
    int* __restrict__ lb, int* __restrict__ vd) {
  int i = blockIdx.x * blockDim.x + threadIdx.x;
  int M = Nroi * (NCLS - 1);
  if (i >= M) return;
  int roi = i / (NCLS - 1);
  int j = i % (NCLS - 1) + 1;  // skip background class 0
  const float* s = cls + (size_t)roi * NCLS;
  float mx = s[0];
  for (int k = 1; k < NCLS; k++) mx = fmaxf(mx, s[k]);
  float den = 0.0f;
  for (int k = 0; k < NCLS; k++) den += expf(s[k] - mx);
  float score = expf(s[j] - mx) / den;

  float px1 = props[roi * 4 + 0], py1 = props[roi * 4 + 1];
  float px2 = props[roi * 4 + 2], py2 = props[roi * 4 + 3];
  float w = px2 - px1, h = py2 - py1;
  float cx = px1 + 0.5f * w, cy = py1 + 0.5f * h;
  const float* d = reg + (size_t)roi * (NCLS * 4) + j * 4;
  float dmax = logf(1000.0f / 16.0f);
  float dx = d[0], dy = d[1];
  float dw = fminf(d[2], dmax), dh = fminf(d[3], dmax);
  float pcx = dx * w + cx, pcy = dy * h + cy;
  float pw = expf(dw) * w, ph = expf(dh) * h;
  float Wimg = (float)iwp[0], Himg = (float)ihp[0];
  float bx1 = fminf(fmaxf(pcx - 0.5f * pw, 0.0f), Wimg);
  float by1 = fminf(fmaxf(pcy - 0.5f * ph, 0.0f), Himg);
  float bx2 = fminf(fmaxf(pcx + 0.5f * pw, 0.0f), Wimg);
  float by2 = fminf(fmaxf(pcy + 0.5f * ph, 0.0f), Himg);
  int valid = (score > 0.05f) && ((bx2 - bx1) >= 16.0f) && ((by2 - by1) >= 16.0f);
  float off = (float)j * 4096.0f;
  obox[i * 4 + 0] = bx1; obox[i * 4 + 1] = by1;
  obox[i * 4 + 2] = bx2; obox[i * 4 + 3] = by2;
  nbox[i * 4 + 0] = bx1 + off; nbox[i * 4 + 1] = by1 + off;
  nbox[i * 4 + 2] = bx2 + off; nbox[i * 4 + 3] = by2 + off;
  sc[i] = score; lb[i] = j; vd[i] = valid;
}

// ---------------- bitonic sort (descending masked score, stable by index) ----------------
__global__ __launch_bounds__(1024) void sort_kernel(
    const float* __restrict__ sc, const int* __restrict__ vd, int M,
    int* __restrict__ sidx) {
  __shared__ unsigned int key[SORTN];
  __shared__ unsigned short idx[SORTN];
  for (int t = threadIdx.x; t < SORTN; t += 1024) {
    unsigned int k = 0xFFFFFFFFu;
    if (t < M) {
      float sm = vd[t] ? sc[t] : -INFINITY;  // == argsort(where(valid,-s,inf))
      unsigned int u = __float_as_uint(sm);
      u = (u & 0x80000000u) ? ~u : (u | 0x80000000u);  // monotone ascending
      k = ~u;                                          // ascending key == descending score
    }
    key[t] = k;
    idx[t] = (unsigned short)t;
  }
  __syncthreads();
  for (int kk = 2; kk <= SORTN; kk <<= 1) {
    for (int jj = kk >> 1; jj > 0; jj >>= 1) {
      for (int t = threadIdx.x; t < SORTN; t += 1024) {
        int p = t ^ jj;
        if (p > t) {
          bool asc = ((t & kk) == 0);
          unsigned int ka = key[t], kb = key[p];
          unsigned short ia = idx[t], ib = idx[p];
          bool gt = (ka > kb) || (ka == kb && ia > ib);
          if (gt == asc) { key[t] = kb; key[p] = ka; idx[t] = ib; idx[p] = ia; }
        }
      }
      __syncthreads();
    }
  }
  for (int t = threadIdx.x; t < SORTN; t += 1024) sidx[t] = (int)idx[t];
}

// ---------------- gather candidates into sorted order ----------------
__global__ __launch_bounds__(256) void gather_kernel(
    const int* __restrict__ sidx, const float* __restrict__ nbox,
    const float* __restrict__ obox, const float* __restrict__ sc,
    const int* __restrict__ lb, const int* __restrict__ vd, int M,
    float* __restrict__ snb, float* __restrict__ sob, float* __restrict__ ssc,
    int* __restrict__ slb, int* __restrict__ svd) {
  int t = blockIdx.x * blockDim.x + threadIdx.x;
  if (t >= M) return;
  int s = sidx[t];
#pragma unroll
  for (int q = 0; q < 4; q++) {
    snb[t * 4 + q] = nbox[s * 4 + q];
    sob[t * 4 + q] = obox[s * 4 + q];
  }
  ssc[t] = sc[s];
  slb[t] = lb[s];
  svd[t] = vd[s];
}

// ---------------- greedy NMS over sorted (class-offset) boxes ----------------
__global__ __launch_bounds__(1024) void nms_kernel(
    const float* __restrict__ snb, const int* __restrict__ svd, int M,
    int* __restrict__ keepg) {
  __shared__ unsigned char keep[SORTN];
  float rx1[NSMAX], ry1[NSMAX], rx2[NSMAX], ry2[NSMAX], rar[NSMAX];
#pragma unroll
  for (int s = 0; s < NSMAX; s++) {
    int j = threadIdx.x + s * 1024;
    if (j < M) {
      rx1[s] = snb[j * 4 + 0]; ry1[s] = snb[j * 4 + 1];
      rx2[s] = snb[j * 4 + 2]; ry2[s] = snb[j * 4 + 3];
      rar[s] = (rx2[s] - rx1[s]) * (ry2[s] - ry1[s]);
      keep[j] = svd[j] ? 1 : 0;
    } else {
      rx1[s] = ry1[s] = rx2[s] = ry2[s] = rar[s] = 0.0f;
    }
  }
  __syncthreads();
  for (int i = 0; i < M; i++) {
    if (keep[i]) {  // uniform across block (same LDS value)
      float ix1 = snb[i * 4 + 0], iy1 = snb[i * 4 + 1];
      float ix2 = snb[i * 4 + 2], iy2 = snb[i * 4 + 3];
      float ia = (ix2 - ix1) * (iy2 - iy1);
#pragma unroll
      for (int s = 0; s < NSMAX; s++) {
        int j = threadIdx.x + s * 1024;
        if (j < M && j > i && keep[j]) {
          float xx1 = fmaxf(ix1, rx1[s]), yy1 = fmaxf(iy1, ry1[s]);
          float xx2 = fminf(ix2, rx2[s]), yy2 = fminf(iy2, ry2[s]);
          float inter = fmaxf(xx2 - xx1, 0.0f) * fmaxf(yy2 - yy1, 0.0f);
          float iou = inter / (ia + rar[s] - inter + 1e-9f);
          if (iou > 0.5f) keep[j] = 0;
        }
      }
      __syncthreads();
    }
  }
  __syncthreads();
  for (int j = threadIdx.x; j < M; j += 1024) keepg[j] = keep[j];
}

// ---------------- top-100 emit (sorted order == top_k of masked scores) ----------------
__global__ void final_kernel(const float* __restrict__ sob,
                             const float* __restrict__ ssc,
                             const int* __restrict__ slb,
                             const int* __restrict__ keepg, int M,
                             float* __restrict__ out) {
  if (threadIdx.x != 0 || blockIdx.x != 0) return;
  int r = 0;
  for (int j = 0; j < M && r < 100; j++) {
    if (keepg[j]) {
      out[r * 4 + 0] = sob[j * 4 + 0];
      out[r * 4 + 1] = sob[j * 4 + 1];
      out[r * 4 + 2] = sob[j * 4 + 2];
      out[r * 4 + 3] = sob[j * 4 + 3];
      out[400 + r] = ssc[j];
      out[500 + r] = (float)slb[j];
      out[600 + r] = 1.0f;
      r++;
    }
  }
  for (; r < 100; r++) {
    out[r * 4 + 0] = 0.0f; out[r * 4 + 1] = 0.0f;
    out[r * 4 + 2] = 0.0f; out[r * 4 + 3] = 0.0f;
    out[400 + r] = 0.0f;
    out[500 + r] = -1.0f;
    out[600 + r] = 0.0f;
  }
}

// ---------------- launch ----------------
static inline size_t align256(size_t x) { return (x + 255) & ~(size_t)255; }

extern "C" void kernel_launch(void* const* d_in, const int* in_sizes, int n_in,
                              void* d_out, int out_size, void* d_ws, size_t ws_size,
                              hipStream_t stream) {
  const float* feat  = (const float*)d_in[0];
  const float* props = (const float*)d_in[1];
  const float* w6    = (const float*)d_in[2];
  const float* b6    = (const float*)d_in[3];
  const float* w7    = (const float*)d_in[4];
  const float* b7    = (const float*)d_in[5];
  const float* wcls  = (const float*)d_in[6];
  const float* bcls  = (const float*)d_in[7];
  const float* wreg  = (const float*)d_in[8];
  const float* breg  = (const float*)d_in[9];
  const int* ihp     = (const int*)d_in[10];
  const int* iwp     = (const int*)d_in[11];

  const int Nroi = in_sizes[1] / 4;                 // 300
  const int Mp = ((Nroi + 31) / 32) * 32;           // 320 (zero-padded rows)
  const int Mcand = Nroi * (NCLS - 1);              // 6000

  // carve workspace
  char* ws = (char*)d_ws;
  size_t off = 0;
  float* X   = (float*)(ws + off); off += align256((size_t)Mp * FDIM * 4);
  float* H6  = (float*)(ws + off); off += align256((size_t)Mp * FC * 4);
  float* H7  = (float*)(ws + off); off += align256((size_t)Mp * FC * 4);
  float* CLS = (float*)(ws + off); off += align256((size_t)Mp * NCLS * 4);
  float* REG = (float*)(ws + off); off += align256((size_t)Mp * NCLS * 4 * 4);
  float* NB  = (float*)(ws + off); off += align256((size_t)SORTN * 4 * 4);
  float* OB  = (float*)(ws + off); off += align256((size_t)SORTN * 4 * 4);
  float* SC  = (float*)(ws + off); off += align256((size_t)SORTN * 4);
  int*   LB  = (int*)(ws + off);   off += align256((size_t)SORTN * 4);
  int*   VD  = (int*)(ws + off);   off += align256((size_t)SORTN * 4);
  int*   SIDX= (int*)(ws + off);   off += align256((size_t)SORTN * 4);
  float* SNB = (float*)(ws + off); off += align256((size_t)SORTN * 4 * 4);
  float* SOB = (float*)(ws + off); off += align256((size_t)SORTN * 4 * 4);
  float* SSC = (float*)(ws + off); off += align256((size_t)SORTN * 4);
  int*   SLB = (int*)(ws + off);   off += align256((size_t)SORTN * 4);
  int*   SVD = (int*)(ws + off);   off += align256((size_t)SORTN * 4);
  int*   KEEP= (int*)(ws + off);   off += align256((size_t)SORTN * 4);

  // 1) ROI pool -> X [Mp, 12544] (rows >= Nroi zeroed)
  roi_pool_kernel<<<Mp, 256, 0, stream>>>(feat, props, ihp, X, Nroi);

  // 2) FC6: X @ w6^T + b6, ReLU  (weight-BW-bound fp32 WMMA, async double-buffered)
  gemm_wmma_kernel<<<dim3(FC / 64, Mp / 32), 256, 0, stream>>>(
      X, w6, b6, H6, FDIM, FC, 1);

  // 3) FC7: H6 @ w7^T + b7, ReLU
  gemm_wmma_kernel<<<dim3(FC / 64, Mp / 32), 256, 0, stream>>>(
      H6, w7, b7, H7, FC, FC, 1);

  // 4) heads: cls [*,21], reg [*,84]
  gemm_wmma_kernel<<<dim3(1, Mp / 32), 256, 0, stream>>>(
      H7, wcls, bcls, CLS, FC, NCLS, 0);
  gemm_wmma_kernel<<<dim3(2, Mp / 32), 256, 0, stream>>>(
      H7, wreg, breg, REG, FC, NCLS * 4, 0);

  // 5) softmax + decode + candidates
  cand_kernel<<<(Mcand + 255) / 256, 256, 0, stream>>>(
      CLS, REG, props, ihp, iwp, Nroi, NB, OB, SC, LB, VD);

  // 6) sort by masked score desc (stable)
  sort_kernel<<<1, 1024, 0, stream>>>(SC, VD, Mcand, SIDX);

  // 7) gather sorted
  gather_kernel<<<(SORTN + 255) / 256, 256, 0, stream>>>(
      SIDX, NB, OB, SC, LB, VD, Mcand, SNB, SOB, SSC, SLB, SVD);

  // 8) greedy NMS
  nms_kernel<<<1, 1024, 0, stream>>>(SNB, SVD, Mcand, KEEP);

  // 9) top-100 emit
  final_kernel<<<1, 32, 0, stream>>>(SOB, SSC, SLB, KEEP, Mcand, (float*)d_out);

  (void)n_in; (void)out_size; (void)ws_size;
}